// RadialAngularEmbedding_50714973831804
// MI455X (gfx1250) — compile-verified
//
#include <hip/hip_runtime.h>
#include <hip/hip_bf16.h>

typedef float v2f __attribute__((ext_vector_type(2)));
typedef float v8f __attribute__((ext_vector_type(8)));

#define C64 64
#define COLS 576   // C * (1+3+5)
#define SHP 68     // padded LDS row stride for h tiles (bank-conflict free)
#define WAVES_A 6  // waves per block in edge kernel

__device__ __forceinline__ float silu_f(float x) {
    // x * 1/(1+e^-x) with hw fast reciprocal (avoids IEEE div expansion)
    return x * __builtin_amdgcn_rcpf(1.0f + __expf(-x));
}

__device__ __forceinline__ v8f wmma_f32(v2f a, v2f b, v8f c) {
    return __builtin_amdgcn_wmma_f32_16x16x4_f32(false, a, false, b, (short)0, c, false, false);
}

// ---------------------------------------------------------------------------
// Kernel 0: zero the segment-sum accumulator (d_ws) and the sc half of d_out
// ---------------------------------------------------------------------------
__global__ void zero_kernel(float4* __restrict__ msg, float4* __restrict__ sc, long n4) {
    long i = (long)blockIdx.x * blockDim.x + threadIdx.x;
    long stride = (long)gridDim.x * blockDim.x;
    const float4 z = make_float4(0.f, 0.f, 0.f, 0.f);
    for (; i < n4; i += stride) {
        msg[i] = z;
        sc[i]  = z;
    }
}

// ---------------------------------------------------------------------------
// Edge-side epilogue: wx outer-product with Y_l, scatter-add into msg
// ---------------------------------------------------------------------------
template <int DL, int BASE, int OFF>
__device__ __forceinline__ void scatter_l(const v8f& accw, int half, int row16, int c,
                                          long e0, int E,
                                          const int* sS, const int* sR, const float* sYw,
                                          const float* __restrict__ nf,
                                          float* __restrict__ msg) {
#pragma unroll
    for (int v = 0; v < 8; ++v) {
        int r = half ? (v + 8) : v;
        long e = e0 + r;
        if (e < E) {
            float wv = accw[v] * 0.125f;                         // W4 output scale
            float wx = wv * nf[(long)sS[r] * C64 + c];           // * node_features[sender]
            float* dst = msg + (long)sR[r] * COLS + BASE + c * DL;
#pragma unroll
            for (int m = 0; m < DL; ++m)
                unsafeAtomicAdd(dst + m, wx * sYw[r * 9 + OFF + m]);
        }
    }
}

// ---------------------------------------------------------------------------
// Kernel 1: edge MLP (WMMA f32 chain) + outer products + atomic segment-sum
//   block = 192 threads = 6 waves, each wave owns a 16-edge tile
// ---------------------------------------------------------------------------
__global__ __launch_bounds__(192) void edge_mlp_scatter(
    const float* __restrict__ length, const float* __restrict__ nf,
    const float* __restrict__ ea,
    const float* __restrict__ W1, const float* __restrict__ W2,
    const float* __restrict__ W3, const float* __restrict__ W4,
    const int* __restrict__ snd, const int* __restrict__ rcv,
    float* __restrict__ msg, int E) {

    __shared__ float sW2[C64 * C64];
    __shared__ float sW3[C64 * C64];
    __shared__ float sh[WAVES_A][16 * SHP];   // per-wave h tile (16 x 64, padded)
    __shared__ float sY[WAVES_A][16 * 9];     // per-wave edge spherical harmonics
    __shared__ int   sS[WAVES_A][16];         // senders
    __shared__ int   sR[WAVES_A][16];         // receivers

    const int tid = threadIdx.x;
    for (int i = tid; i < C64 * C64; i += 192) {
        sW2[i] = W2[i];
        sW3[i] = W3[i];
    }
    __syncthreads();

    const int wid = tid >> 5;
    const int lane = tid & 31;
    const int row16 = lane & 15;
    const int half = lane >> 4;
    const long e0 = ((long)blockIdx.x * WAVES_A + wid) * 16;

    // per-wave edge metadata (cross-lane reads later are in-order LDS within the wave)
    if (lane < 16) {
        long e = e0 + lane;
        if (e >= E) e = E - 1;
        sS[wid][lane] = snd[e];
        sR[wid][lane] = rcv[e];
#pragma unroll
        for (int q = 0; q < 9; ++q) sY[wid][lane * 9 + q] = ea[e * 9 + q];
    }

    long erow = e0 + row16;
    if (erow >= E) erow = E - 1;

    v8f acc[4];

    // ---- layer 1: (16x8) @ W1(8x64), scale 1/sqrt(8), SiLU -> sh
    {
#pragma unroll
        for (int j = 0; j < 4; ++j) acc[j] = (v8f){};
#pragma unroll
        for (int kk = 0; kk < 2; ++kk) {
            const int kb = kk * 4 + half * 2;
            v2f a;
            a.x = length[erow * 8 + kb];
            a.y = length[erow * 8 + kb + 1];
#pragma unroll
            for (int j = 0; j < 4; ++j) {
                const int col = j * 16 + row16;
                v2f b;
                b.x = W1[kb * C64 + col];
                b.y = W1[(kb + 1) * C64 + col];
                acc[j] = wmma_f32(a, b, acc[j]);
            }
        }
        const float s1 = 0.35355339059327373f;  // 1/sqrt(8)
#pragma unroll
        for (int j = 0; j < 4; ++j)
#pragma unroll
            for (int v = 0; v < 8; ++v) {
                const int r = half ? (v + 8) : v;
                sh[wid][r * SHP + j * 16 + row16] = silu_f(acc[j][v] * s1);
            }
    }

    // ---- layers 2,3: (16x64) @ W(64x64), scale 0.125, SiLU, in-place into sh
#pragma unroll 1
    for (int layer = 0; layer < 2; ++layer) {
        const float* W = (layer == 0) ? sW2 : sW3;  // no LDS pointer table (addrspacecast init)
#pragma unroll
        for (int j = 0; j < 4; ++j) acc[j] = (v8f){};
#pragma unroll 1
        for (int kk = 0; kk < 16; ++kk) {
            const int kb = kk * 4 + half * 2;
            v2f a;
            a.x = sh[wid][row16 * SHP + kb];
            a.y = sh[wid][row16 * SHP + kb + 1];
#pragma unroll
            for (int j = 0; j < 4; ++j) {
                const int col = j * 16 + row16;
                v2f b;
                b.x = W[kb * C64 + col];
                b.y = W[(kb + 1) * C64 + col];
                acc[j] = wmma_f32(a, b, acc[j]);
            }
        }
        // all reads of sh done (accs live in regs); per-wave LDS ops are in-order
#pragma unroll
        for (int j = 0; j < 4; ++j)
#pragma unroll
            for (int v = 0; v < 8; ++v) {
                const int r = half ? (v + 8) : v;
                sh[wid][r * SHP + j * 16 + row16] = silu_f(acc[j][v] * 0.125f);
            }
    }

    // ---- layer 4: (16x64) @ W4(64x192), scale 0.125 (no SiLU) + fused scatter
    auto l4acc = [&](int j) -> v8f {
        v8f a4 = (v8f){};
#pragma unroll 1
        for (int kk = 0; kk < 16; ++kk) {
            const int kb = kk * 4 + half * 2;
            v2f a;
            a.x = sh[wid][row16 * SHP + kb];
            a.y = sh[wid][row16 * SHP + kb + 1];
            const int col = j * 16 + row16;
            v2f b;
            b.x = W4[kb * 192 + col];
            b.y = W4[(kb + 1) * 192 + col];
            a4 = wmma_f32(a, b, a4);
        }
        return a4;
    };

    // l = 0 (d=1, base 0, Y offset 0): j = 0..3
#pragma unroll 1
    for (int jj = 0; jj < 4; ++jj) {
        v8f aw = l4acc(jj);
        const int c = jj * 16 + row16;
        scatter_l<1, 0, 0>(aw, half, row16, c, e0, E, sS[wid], sR[wid], sY[wid], nf, msg);
    }
    // l = 1 (d=3, base 64, Y offset 1): j = 4..7
#pragma unroll 1
    for (int jj = 0; jj < 4; ++jj) {
        v8f aw = l4acc(4 + jj);
        const int c = jj * 16 + row16;
        scatter_l<3, 64, 1>(aw, half, row16, c, e0, E, sS[wid], sR[wid], sY[wid], nf, msg);
    }
    // l = 2 (d=5, base 256, Y offset 4): j = 8..11
#pragma unroll 1
    for (int jj = 0; jj < 4; ++jj) {
        v8f aw = l4acc(8 + jj);
        const int c = jj * 16 + row16;
        scatter_l<5, 256, 4>(aw, half, row16, c, e0, E, sS[wid], sR[wid], sY[wid], nf, msg);
    }
}

// ---------------------------------------------------------------------------
// Kernel 2: node-side channel mixing (Wl) + bilinear self-connection (Wsc)
//   block = 256 threads = 8 waves, each wave owns a 16-node tile
// ---------------------------------------------------------------------------
__global__ __launch_bounds__(256) void node_mix(
    const float* __restrict__ msg, const float* __restrict__ nf,
    const float* __restrict__ na, const float* __restrict__ Wl,
    const float* __restrict__ Wsc,
    float* __restrict__ out_msg, float* __restrict__ out_sc, int N) {

    const int tid = threadIdx.x;
    const int wid = tid >> 5;
    const int lane = tid & 31;
    const int row16 = lane & 15;
    const int half = lane >> 4;
    const long n0 = ((long)blockIdx.x * 8 + wid) * 16;
    long nA = n0 + row16;
    if (nA >= N) nA = N - 1;  // clamped A-frag row; stores below are guarded

    // ---- message: out[n,v,mg] = (1/8) * sum_u msg[n, base + u*dl + m] * Wl[l][u][v]
    const int DLs[3]   = { 1, 3, 5 };
    const int BASEs[3] = { 0, 64, 256 };
    const int MGs[3]   = { 0, 1, 4 };
#pragma unroll 1
    for (int l = 0; l < 3; ++l) {
        const int dl = DLs[l], base = BASEs[l];
#pragma unroll 1
        for (int m = 0; m < dl; ++m) {
            const int mg = MGs[l] + m;
#pragma unroll 1
            for (int j = 0; j < 4; ++j) {
                v8f acc = (v8f){};
#pragma unroll 1
                for (int kk = 0; kk < 16; ++kk) {
                    const int kb = kk * 4 + half * 2;
                    v2f a;
                    a.x = msg[nA * COLS + base + kb * dl + m];
                    a.y = msg[nA * COLS + base + (kb + 1) * dl + m];
                    const int col = j * 16 + row16;
                    v2f b;
                    b.x = Wl[(l * C64 + kb) * C64 + col];
                    b.y = Wl[(l * C64 + kb + 1) * C64 + col];
                    acc = wmma_f32(a, b, acc);
                }
#pragma unroll
                for (int v = 0; v < 8; ++v) {
                    const int r = half ? (v + 8) : v;
                    const long n = n0 + r;
                    if (n < N)
                        out_msg[n * COLS + (long)(j * 16 + row16) * 9 + mg] = acc[v] * 0.125f;
                }
            }
        }
    }

    // ---- sc0: fold na[n,a] into A rows, accumulate over a and u in one WMMA chain
    float navals[4];
#pragma unroll
    for (int a = 0; a < 4; ++a) navals[a] = na[nA * 4 + a];

#pragma unroll 1
    for (int j = 0; j < 4; ++j) {
        v8f acc = (v8f){};
#pragma unroll 1
        for (int aix = 0; aix < 4; ++aix) {
            const float sA = navals[aix];
#pragma unroll 1
            for (int kk = 0; kk < 16; ++kk) {
                const int kb = kk * 4 + half * 2;
                v2f a;
                a.x = nf[nA * C64 + kb] * sA;
                a.y = nf[nA * C64 + kb + 1] * sA;
                const int col = j * 16 + row16;
                v2f b;
                b.x = Wsc[(aix * C64 + kb) * C64 + col];
                b.y = Wsc[(aix * C64 + kb + 1) * C64 + col];
                acc = wmma_f32(a, b, acc);
            }
        }
#pragma unroll
        for (int v = 0; v < 8; ++v) {
            const int r = half ? (v + 8) : v;
            const long n = n0 + r;
            if (n < N)
                out_sc[n * COLS + j * 16 + row16] = acc[v] * 0.0625f;  // 1/sqrt(NA*C)
        }
    }
}

// ---------------------------------------------------------------------------
extern "C" void kernel_launch(void* const* d_in, const int* in_sizes, int n_in,
                              void* d_out, int out_size, void* d_ws, size_t ws_size,
                              hipStream_t stream) {
    const float* length = (const float*)d_in[0];
    const float* nf     = (const float*)d_in[1];
    const float* na     = (const float*)d_in[2];
    const float* ea     = (const float*)d_in[3];
    const float* W1     = (const float*)d_in[4];
    const float* W2     = (const float*)d_in[5];
    const float* W3     = (const float*)d_in[6];
    const float* W4     = (const float*)d_in[7];
    const float* Wl     = (const float*)d_in[8];
    const float* Wsc    = (const float*)d_in[9];
    const int*   eidx   = (const int*)d_in[10];

    const int E = in_sizes[0] / 8;    // 320000
    const int N = in_sizes[1] / C64;  // 20000

    float* msg     = (float*)d_ws;                 // N x 576 accumulator
    float* out_msg = (float*)d_out;                // message, N x 576
    float* out_sc  = out_msg + (long)N * COLS;     // sc, N x 576

    const long acc_elems4 = (long)N * COLS / 4;    // COLS divisible by 4
    zero_kernel<<<512, 256, 0, stream>>>((float4*)msg, (float4*)out_sc, acc_elems4);

    const int edge_blocks = (E + WAVES_A * 16 - 1) / (WAVES_A * 16);
    edge_mlp_scatter<<<edge_blocks, 192, 0, stream>>>(
        length, nf, ea, W1, W2, W3, W4, eidx, eidx + E, msg, E);

    const int node_blocks = (N + 8 * 16 - 1) / (8 * 16);
    node_mix<<<node_blocks, 256, 0, stream>>>(msg, nf, na, Wl, Wsc, out_msg, out_sc, N);
}